// DecoderRNN_83554293776653
// MI455X (gfx1250) — compile-verified
//
#include <hip/hip_runtime.h>
#include <hip/hip_bf16.h>

#define B_      256
#define S_      200
#define V_      32000
#define E_      512
#define DEC_H_  512
#define ENC_D_  1024
#define COMB_   2048
#define RNN_IN_ 1536
#define G3_     1536
#define INF_    1000000.0f

typedef __attribute__((ext_vector_type(2))) float v2f;
typedef __attribute__((ext_vector_type(4))) float v4f;
typedef __attribute__((ext_vector_type(8))) float v8f;
typedef __attribute__((ext_vector_type(4))) int   v4i;

// ---------------------------------------------------------------------------
// CDNA5 async global->LDS staging (GLOBAL_LOAD_ASYNC_TO_LDS_B128, ASYNCcnt).
// Builtin signature (from hipcc diagnostic): takes int4 addrspace(1)* src and
// an LDS destination pointer, plus immediate offset + cpol.
// Falls back to a synchronous 16B copy if the builtin is unavailable.
// ---------------------------------------------------------------------------
#if __has_builtin(__builtin_amdgcn_global_load_async_to_lds_b128)
#define HAVE_ASYNC_LDS 1
#else
#define HAVE_ASYNC_LDS 0
#endif

__device__ __forceinline__ void stage16_async(const float* gsrc, float* ldst)
{
#if HAVE_ASYNC_LDS
    __builtin_amdgcn_global_load_async_to_lds_b128(
        (__attribute__((address_space(1))) v4i*)gsrc,
        (__attribute__((address_space(3))) v4i*)ldst,
        /*imm_offset=*/0, /*cpol=*/0);
#else
    *reinterpret_cast<v4f*>(ldst) = *reinterpret_cast<const v4f*>(gsrc);
#endif
}

__device__ __forceinline__ void wait_async_done()
{
#if HAVE_ASYNC_LDS
#if __has_builtin(__builtin_amdgcn_s_wait_asynccnt)
    __builtin_amdgcn_s_wait_asynccnt(0);
#else
    asm volatile("s_wait_asynccnt 0" ::: "memory");
#endif
#endif
}

// ---------------------------------------------------------------------------
// Generic fp32 WMMA GEMM:  C[M,N] = A[M,K] @ W[N,K]^T (+ bias[N])
// M == 256 always. One block = 16-wide N strip, 8 waves; each wave owns two
// 16x16 M-tiles. W tile (16 rows x 64 K) double-buffered in LDS via async
// DMA so the fc_w HBM stream overlaps the V_WMMA_F32_16X16X4_F32 compute.
// K must be a multiple of 64.
// ---------------------------------------------------------------------------
__global__ __launch_bounds__(256) void gemm_wmma_f32(
    const float* __restrict__ A, const float* __restrict__ W,
    const float* __restrict__ bias, float* __restrict__ C,
    int M, int N, int K)
{
    __shared__ float wt[2][16 * 64];

    const int tid  = threadIdx.x;
    const int wave = tid >> 5;
    const int lane = tid & 31;
    const int half = lane >> 4;     // 0: lanes 0-15, 1: lanes 16-31
    const int l    = lane & 15;
    const int n0   = blockIdx.x * 16;

    // staging assignment: thread -> one 16B slot of the 16x64 tile
    const int idx = tid * 4;
    const int row = idx >> 6;
    const int col = idx & 63;
    const float* wbase = W + (size_t)(n0 + row) * K + col;

    v8f acc0, acc1;
    #pragma unroll
    for (int r = 0; r < 8; ++r) { acc0[r] = 0.0f; acc1[r] = 0.0f; }

    const int nchunks = K >> 6;          // K/64

    // prologue: stage chunk 0 into buffer 0
    stage16_async(wbase, &wt[0][idx]);

    for (int i = 0; i < nchunks; ++i) {
        // chunk i's DMA (issued one compute-phase ago) must be complete,
        // and all waves must be past reading the buffer we re-target next.
        wait_async_done();
        __syncthreads();

        if (i + 1 < nchunks)
            stage16_async(wbase + (i + 1) * 64, &wt[(i + 1) & 1][idx]);

        const float* wcur  = &wt[i & 1][0];
        const float* arow0 = A + (size_t)(wave * 16 + l) * K + i * 64 + 2 * half;
        const float* arow1 = arow0 + (size_t)128 * K;   // M-tile wave+8

        #pragma unroll
        for (int k = 0; k < 64; k += 4) {
            v2f bf = *reinterpret_cast<const v2f*>(wcur + l * 64 + k + 2 * half);
            v2f a0 = *reinterpret_cast<const v2f*>(arow0 + k);
            v2f a1 = *reinterpret_cast<const v2f*>(arow1 + k);
            acc0 = __builtin_amdgcn_wmma_f32_16x16x4_f32(
                       false, a0, false, bf, (short)0, acc0, false, false);
            acc1 = __builtin_amdgcn_wmma_f32_16x16x4_f32(
                       false, a1, false, bf, (short)0, acc1, false, false);
        }
    }

    const float bv = bias ? bias[n0 + l] : 0.0f;
    #pragma unroll
    for (int r = 0; r < 8; ++r) {
        int m0 = wave * 16 + r + 8 * half;
        int m1 = m0 + 128;
        C[(size_t)m0 * N + n0 + l] = acc0[r] + bv;
        C[(size_t)m1 * N + n0 + l] = acc1[r] + bv;
    }
}

// ---------------------------------------------------------------------------
// Embedding gather: emb[word[b]] -> x[b, 0:512] and comb[b, 1536:2048]
// ---------------------------------------------------------------------------
__global__ __launch_bounds__(128) void embed_gather(
    const int* __restrict__ word, const float* __restrict__ emb,
    float* __restrict__ x, float* __restrict__ comb)
{
    const int b = blockIdx.x, tid = threadIdx.x;
    const int row = word[b];
    v4f v = *reinterpret_cast<const v4f*>(emb + (size_t)row * E_ + tid * 4);
    *reinterpret_cast<v4f*>(x    + (size_t)b * RNN_IN_ + tid * 4)         = v;
    *reinterpret_cast<v4f*>(comb + (size_t)b * COMB_ + RNN_IN_ + tid * 4) = v;
}

// ---------------------------------------------------------------------------
// Fused attention per batch row: att = q . enc, mask PAD, softmax, context.
// ---------------------------------------------------------------------------
__global__ __launch_bounds__(256) void attn_fused(
    const float* __restrict__ q, const float* __restrict__ enc,
    const int* __restrict__ src, float* __restrict__ scores_out,
    float* __restrict__ x, float* __restrict__ comb)
{
    __shared__ float qs[ENC_D_];
    __shared__ float att[S_];
    __shared__ float red[256];

    const int b = blockIdx.x, tid = threadIdx.x;
    const int wave = tid >> 5, lane = tid & 31;

    for (int d = tid; d < ENC_D_; d += 256) qs[d] = q[(size_t)b * ENC_D_ + d];
    __syncthreads();

    const float* eb = enc + (size_t)b * S_ * ENC_D_;
    for (int s = wave; s < S_; s += 8) {
        const float* es = eb + (size_t)s * ENC_D_;
        float p = 0.0f;
        for (int d = lane; d < ENC_D_; d += 32) p += qs[d] * es[d];
        #pragma unroll
        for (int off = 16; off > 0; off >>= 1) p += __shfl_xor(p, off, 32);
        if (lane == 0)
            att[s] = (src[(size_t)b * S_ + s] == 0) ? -INF_ : p;
    }
    __syncthreads();

    float m = -3.4e38f;
    for (int s = tid; s < S_; s += 256) m = fmaxf(m, att[s]);
    red[tid] = m; __syncthreads();
    for (int o = 128; o > 0; o >>= 1) {
        if (tid < o) red[tid] = fmaxf(red[tid], red[tid + o]);
        __syncthreads();
    }
    m = red[0]; __syncthreads();

    float sum = 0.0f;
    for (int s = tid; s < S_; s += 256) {
        float e = __expf(att[s] - m);
        att[s] = e;
        sum += e;
    }
    red[tid] = sum; __syncthreads();
    for (int o = 128; o > 0; o >>= 1) {
        if (tid < o) red[tid] += red[tid + o];
        __syncthreads();
    }
    const float inv = 1.0f / red[0];

    for (int s = tid; s < S_; s += 256) {
        float sc = att[s] * inv;
        att[s] = sc;
        scores_out[(size_t)b * S_ + s] = sc;
    }
    __syncthreads();

    for (int d = tid; d < ENC_D_; d += 256) {
        float c = 0.0f;
        for (int s = 0; s < S_; ++s) c += att[s] * eb[(size_t)s * ENC_D_ + d];
        x[(size_t)b * RNN_IN_ + E_ + d]      = c;
        comb[(size_t)b * COMB_ + DEC_H_ + d] = c;
    }
}

// ---------------------------------------------------------------------------
// GRU gates -> hidden (into comb[:, :512] and d_out hidden slot)
// ---------------------------------------------------------------------------
__global__ __launch_bounds__(256) void gru_gate(
    const float* __restrict__ gi, const float* __restrict__ gh,
    const float* __restrict__ h_prev, float* __restrict__ comb,
    float* __restrict__ h_out)
{
    const int idx = blockIdx.x * 256 + threadIdx.x;
    if (idx >= B_ * DEC_H_) return;
    const int b = idx / DEC_H_, j = idx % DEC_H_;
    const float* gib = gi + (size_t)b * G3_;
    const float* ghb = gh + (size_t)b * G3_;
    float r = 1.0f / (1.0f + __expf(-(gib[j] + ghb[j])));
    float z = 1.0f / (1.0f + __expf(-(gib[DEC_H_ + j] + ghb[DEC_H_ + j])));
    float n = tanhf(gib[2 * DEC_H_ + j] + r * ghb[2 * DEC_H_ + j]);
    float h = (1.0f - z) * n + z * h_prev[idx];
    comb[(size_t)b * COMB_ + j] = h;
    h_out[idx] = h;
}

// ---------------------------------------------------------------------------
// Per-row softmax stats over V + pointer-gate sigmoid.
// ---------------------------------------------------------------------------
__global__ __launch_bounds__(256) void row_stats(
    const float* __restrict__ logits, const float* __restrict__ comb,
    const float* __restrict__ ptr_w, const float* __restrict__ ptr_b,
    float* __restrict__ mrow, float* __restrict__ srow, float* __restrict__ pg)
{
    __shared__ float red[256];
    const int b = blockIdx.x, tid = threadIdx.x;
    const float* lb = logits + (size_t)b * V_;

    float m = -3.4e38f;
    for (int v = tid; v < V_; v += 256) m = fmaxf(m, lb[v]);
    red[tid] = m; __syncthreads();
    for (int o = 128; o > 0; o >>= 1) {
        if (tid < o) red[tid] = fmaxf(red[tid], red[tid + o]);
        __syncthreads();
    }
    m = red[0]; __syncthreads();

    float s = 0.0f;
    for (int v = tid; v < V_; v += 256) s += __expf(lb[v] - m);
    red[tid] = s; __syncthreads();
    for (int o = 128; o > 0; o >>= 1) {
        if (tid < o) red[tid] += red[tid + o];
        __syncthreads();
    }
    s = red[0]; __syncthreads();

    const float* cb = comb + (size_t)b * COMB_;
    float d = 0.0f;
    for (int k = tid; k < COMB_; k += 256) d += cb[k] * ptr_w[k];
    red[tid] = d; __syncthreads();
    for (int o = 128; o > 0; o >>= 1) {
        if (tid < o) red[tid] += red[tid + o];
        __syncthreads();
    }
    if (tid == 0) {
        mrow[b] = m;
        srow[b] = s;
        pg[b]   = 1.0f / (1.0f + __expf(-(red[0] + ptr_b[0])));
    }
}

__global__ __launch_bounds__(256) void finalize_out(
    const float* __restrict__ logits, const float* __restrict__ mrow,
    const float* __restrict__ srow, const float* __restrict__ pg,
    float* __restrict__ out)
{
    const size_t idx = (size_t)blockIdx.x * 256 + threadIdx.x;
    const int b = (int)(idx / V_);
    out[idx] = pg[b] * __expf(logits[idx] - mrow[b]) / srow[b];
}

__global__ __launch_bounds__(256) void scatter_ptr(
    const float* __restrict__ scores, const int* __restrict__ src,
    const float* __restrict__ pg, float* __restrict__ out)
{
    const int idx = blockIdx.x * 256 + threadIdx.x;
    if (idx >= B_ * S_) return;
    const int b = idx / S_;
    atomicAdd(out + (size_t)b * V_ + src[idx], (1.0f - pg[b]) * scores[idx]);
}

// ---------------------------------------------------------------------------
extern "C" void kernel_launch(void* const* d_in, const int* in_sizes, int n_in,
                              void* d_out, int out_size, void* d_ws, size_t ws_size,
                              hipStream_t stream)
{
    (void)in_sizes; (void)n_in; (void)out_size; (void)ws_size;

    const int*   word   = (const int*)  d_in[0];
    const float* h_prev = (const float*)d_in[1];
    const float* enc    = (const float*)d_in[2];
    const int*   src    = (const int*)  d_in[3];
    const float* emb    = (const float*)d_in[4];
    const float* W_attn = (const float*)d_in[5];
    const float* w_ih   = (const float*)d_in[6];
    const float* w_hh   = (const float*)d_in[7];
    const float* b_ih   = (const float*)d_in[8];
    const float* b_hh   = (const float*)d_in[9];
    const float* fc_w   = (const float*)d_in[10];
    const float* fc_b   = (const float*)d_in[11];
    const float* ptr_w  = (const float*)d_in[12];
    const float* ptr_b  = (const float*)d_in[13];

    float* out        = (float*)d_out;                   // B*V
    float* h_out      = out + (size_t)B_ * V_;           // B*512
    float* scores_out = h_out + (size_t)B_ * DEC_H_;     // B*200

    float* ws = (float*)d_ws;
    size_t o = 0;
    float* x      = ws + o; o += (size_t)B_ * RNN_IN_;
    float* comb   = ws + o; o += (size_t)B_ * COMB_;
    float* q      = ws + o; o += (size_t)B_ * ENC_D_;
    float* gi     = ws + o; o += (size_t)B_ * G3_;
    float* gh     = ws + o; o += (size_t)B_ * G3_;
    float* logits = ws + o; o += (size_t)B_ * V_;
    float* mrow   = ws + o; o += B_;
    float* srow   = ws + o; o += B_;
    float* pg     = ws + o; o += B_;

    embed_gather<<<B_, 128, 0, stream>>>(word, emb, x, comb);
    gemm_wmma_f32<<<ENC_D_ / 16, 256, 0, stream>>>(h_prev, W_attn, nullptr, q,
                                                   B_, ENC_D_, DEC_H_);
    attn_fused<<<B_, 256, 0, stream>>>(q, enc, src, scores_out, x, comb);
    gemm_wmma_f32<<<G3_ / 16, 256, 0, stream>>>(x, w_ih, b_ih, gi,
                                                B_, G3_, RNN_IN_);
    gemm_wmma_f32<<<G3_ / 16, 256, 0, stream>>>(h_prev, w_hh, b_hh, gh,
                                                B_, G3_, DEC_H_);
    gru_gate<<<(B_ * DEC_H_) / 256, 256, 0, stream>>>(gi, gh, h_prev, comb, h_out);
    gemm_wmma_f32<<<V_ / 16, 256, 0, stream>>>(comb, fc_w, fc_b, logits,
                                               B_, V_, COMB_);
    row_stats<<<B_, 256, 0, stream>>>(logits, comb, ptr_w, ptr_b, mrow, srow, pg);
    finalize_out<<<(B_ * V_) / 256, 256, 0, stream>>>(logits, mrow, srow, pg, out);
    scatter_ptr<<<(B_ * S_) / 256, 256, 0, stream>>>(scores_out, src, pg, out);
}